// CTR_69355131895965
// MI455X (gfx1250) — compile-verified
//
#include <hip/hip_runtime.h>
#include <math.h>

typedef __attribute__((ext_vector_type(2))) float v2f;
typedef __attribute__((ext_vector_type(8))) float v8f;

#define T_STEPS 8192
#define LSEQ    12
#define HID     64
#define G4H     256

__device__ __forceinline__ float sigf(float x) { return 1.0f / (1.0f + __expf(-x)); }

// ---------------------------------------------------------------------------
// Stage 1: input projection GEMM with fp32 WMMA.
//   G[M,256] = X[M,K] @ W[256,K]^T + b[256]
// One wave per 16(M) x 64(N) macro-tile: A fragments for the whole K loop are
// preloaded into registers (K/4 x v2f), then each k-step does 4 B-fragment
// loads + 4 V_WMMA_F32_16X16X4_F32 into 4 accumulators. K is a template
// parameter so everything fully unrolls and the scheduler can hoist B loads
// across WMMAs instead of s_wait_loadcnt-stalling per instruction.
// Fragment layouts per CDNA5 ISA 7.12.2:
//   A 16x4 : lanes 0-15 -> M=lane, VGPR{0,1}=K{0,1}; lanes 16-31 -> K{2,3}
//   B 4x16 : lanes 0-15 -> N=lane, VGPR{0,1}=K{0,1}; lanes 16-31 -> K{2,3}
//   C 16x16: VGPR r -> M = r + 8*(lane>=16), N = lane&15
// ---------------------------------------------------------------------------
template <int K>
__global__ void proj_wmma_f32(const float* __restrict__ X,
                              const float* __restrict__ W,
                              const float* __restrict__ bias,
                              float* __restrict__ G, int M) {
  constexpr int NK = K / 4;          // full WMMA k-steps (32 for K=128, 16 for K=65)
  const int lane  = threadIdx.x;     // 0..31, full wave, EXEC all ones
  const int m0    = blockIdx.x * 16;
  const int nbase = blockIdx.y * 64; // 4 consecutive 16-wide N tiles
  const int col   = lane & 15;
  const int hi    = lane >> 4;       // 0 or 1

  v8f acc[4];
#pragma unroll
  for (int ti = 0; ti < 4; ++ti) {
    const float bv = bias[nbase + ti * 16 + col];
#pragma unroll
    for (int r = 0; r < 8; ++r) acc[ti][r] = bv;
  }

  // Preload every A fragment for the K loop (per-lane row m = m0 + col).
  const float* Xr = X + (long)(m0 + col) * K;
  v2f afrag[NK];
#pragma unroll
  for (int ks = 0; ks < NK; ++ks) {
    afrag[ks][0] = Xr[4 * ks + 2 * hi];
    afrag[ks][1] = Xr[4 * ks + 2 * hi + 1];
  }

  const float* Wr0 = W + (long)(nbase + col) * K;
  const float* Wr1 = Wr0 + (long)16 * K;
  const float* Wr2 = Wr0 + (long)32 * K;
  const float* Wr3 = Wr0 + (long)48 * K;

#pragma unroll
  for (int ks = 0; ks < NK; ++ks) {
    const int kb = 4 * ks + 2 * hi;
    v2f b0, b1, b2, b3;
    b0[0] = Wr0[kb]; b0[1] = Wr0[kb + 1];
    b1[0] = Wr1[kb]; b1[1] = Wr1[kb + 1];
    b2[0] = Wr2[kb]; b2[1] = Wr2[kb + 1];
    b3[0] = Wr3[kb]; b3[1] = Wr3[kb + 1];
    acc[0] = __builtin_amdgcn_wmma_f32_16x16x4_f32(false, afrag[ks], false, b0,
                                                   (short)0, acc[0], false, false);
    acc[1] = __builtin_amdgcn_wmma_f32_16x16x4_f32(false, afrag[ks], false, b1,
                                                   (short)0, acc[1], false, false);
    acc[2] = __builtin_amdgcn_wmma_f32_16x16x4_f32(false, afrag[ks], false, b2,
                                                   (short)0, acc[2], false, false);
    acc[3] = __builtin_amdgcn_wmma_f32_16x16x4_f32(false, afrag[ks], false, b3,
                                                   (short)0, acc[3], false, false);
  }

  // Scalar K-tail (K=65 -> one column), uniform across the wave.
  if (K & 3) {
#pragma unroll
    for (int k = NK * 4; k < K; ++k) {
#pragma unroll
      for (int ti = 0; ti < 4; ++ti) {
        const float wv = W[(long)(nbase + ti * 16 + col) * K + k];
#pragma unroll
        for (int r = 0; r < 8; ++r)
          acc[ti][r] += X[(long)(m0 + r + 8 * hi) * K + k] * wv;
      }
    }
  }

#pragma unroll
  for (int ti = 0; ti < 4; ++ti)
#pragma unroll
    for (int r = 0; r < 8; ++r)
      G[(long)(m0 + r + 8 * hi) * G4H + nbase + ti * 16 + col] = acc[ti][r];
}

// ---------------------------------------------------------------------------
// Stage 2: serial LSTM recurrences. One block per stream (5 blocks, 256 thr).
// Thread t owns Whh row t (64 VGPRs). Per step: g[t] += Whh[t,:]·h (h in LDS),
// then threads 0..63 run the gate math and update (h, c).
// The precomputed gate input for step t+1 is loaded BEFORE the barriers of
// step t (software pipeline), hiding the global-load latency inside the
// serial chain. Ragged streams additionally prefetch the next G row.
// ---------------------------------------------------------------------------
__global__ void lstm_recur(const float* __restrict__ GvA, const float* __restrict__ GvB,
                           const float* __restrict__ Gim,
                           const float* __restrict__ GlA, const float* __restrict__ GlB,
                           const float* __restrict__ Whh_vad,
                           const float* __restrict__ Whh_img,
                           const float* __restrict__ Whh_lng,
                           const int* __restrict__ lenA, const int* __restrict__ lenB,
                           float* __restrict__ hA, float* __restrict__ hB,
                           float* __restrict__ hI, float* __restrict__ hpa,
                           float* __restrict__ hpb) {
  __shared__ float h_s[HID];
  __shared__ float g_s[G4H];
  const int tid = threadIdx.x;
  const int sid = blockIdx.x;

  const float* G;  const float* Whh;  float* out;  const int* lens = nullptr;
  if      (sid == 0) { G = GvA; Whh = Whh_vad; out = hA; }
  else if (sid == 1) { G = GvB; Whh = Whh_vad; out = hB; }
  else if (sid == 2) { G = Gim; Whh = Whh_img; out = hI; }
  else if (sid == 3) { G = GlA; Whh = Whh_lng; out = hpa; lens = lenA; }
  else               { G = GlB; Whh = Whh_lng; out = hpb; lens = lenB; }

  float w[HID];
#pragma unroll
  for (int k = 0; k < HID; ++k) w[k] = Whh[tid * HID + k];

  if (tid < HID) h_s[tid] = 0.0f;
  float c = 0.0f, prev = 0.0f;
  __syncthreads();

  if (lens == nullptr) {
    // dense stream: one cell per timestep, emit h each step
    float gin = G[tid];                       // pipelined gate-input for step t
    for (int t = 0; t < T_STEPS; ++t) {
      float g = gin;
      if (t + 1 < T_STEPS) gin = G[(long)(t + 1) * G4H + tid];  // issue early
#pragma unroll 8
      for (int k = 0; k < HID; ++k) g += w[k] * h_s[k];
      g_s[tid] = g;
      __syncthreads();
      if (tid < HID) {
        const float gi = g_s[tid], gf = g_s[HID + tid];
        const float gg = g_s[2 * HID + tid], go = g_s[3 * HID + tid];
        c = sigf(gf) * c + sigf(gi) * tanhf(gg);
        const float hn = sigf(go) * tanhf(c);
        h_s[tid] = hn;
        out[(long)t * HID + tid] = hn;
      }
      __syncthreads();
    }
  } else {
    // ragged stream: len[t] inner cells; state persists; empty step emits prev
    for (int t = 0; t < T_STEPS; ++t) {
      int n = lens[t];
      if (n > LSEQ) n = LSEQ;
      if (n < 0)    n = 0;
      for (int j = 0; j < n; ++j) {
        const long row = (long)t * LSEQ + j;
        float g = G[row * G4H + tid];
        __builtin_prefetch(&G[(row + 1) * G4H + tid], 0, 3);  // warm next row
#pragma unroll 8
        for (int k = 0; k < HID; ++k) g += w[k] * h_s[k];
        g_s[tid] = g;
        __syncthreads();
        if (tid < HID) {
          const float gi = g_s[tid], gf = g_s[HID + tid];
          const float gg = g_s[2 * HID + tid], go = g_s[3 * HID + tid];
          c = sigf(gf) * c + sigf(gi) * tanhf(gg);
          h_s[tid] = sigf(go) * tanhf(c);
        }
        __syncthreads();
      }
      if (tid < HID) {
        const float hp = (n > 0) ? h_s[tid] : prev;
        prev = hp;
        out[(long)t * HID + tid] = hp;
      }
      if (t + 1 < T_STEPS)
        __builtin_prefetch(&G[((long)(t + 1) * LSEQ) * G4H + tid], 0, 3);
      __syncthreads();
    }
  }
}

// ---------------------------------------------------------------------------
// Stage 3: alpha = sigmoid(concat(hA,hB,hI,hpa,hpb) @ fc_w^T + fc_b)
// ---------------------------------------------------------------------------
__global__ void fc_sigmoid(const float* __restrict__ hA, const float* __restrict__ hB,
                           const float* __restrict__ hI, const float* __restrict__ hpa,
                           const float* __restrict__ hpb,
                           const float* __restrict__ fcw, const float* __restrict__ fcb,
                           float* __restrict__ out) {
  __shared__ float w_s[5 * HID];
  const int tid = threadIdx.x;
  for (int k = tid; k < 5 * HID; k += blockDim.x) w_s[k] = fcw[k];
  __syncthreads();
  const int t = blockIdx.x * blockDim.x + tid;
  if (t < T_STEPS) {
    float s = fcb[0];
#pragma unroll 8
    for (int k = 0; k < HID; ++k) {
      s += w_s[k]           * hA [(long)t * HID + k];
      s += w_s[HID + k]     * hB [(long)t * HID + k];
      s += w_s[2 * HID + k] * hI [(long)t * HID + k];
      s += w_s[3 * HID + k] * hpa[(long)t * HID + k];
      s += w_s[4 * HID + k] * hpb[(long)t * HID + k];
    }
    out[t] = 1.0f / (1.0f + __expf(-s));
  }
}

extern "C" void kernel_launch(void* const* d_in, const int* in_sizes, int n_in,
                              void* d_out, int out_size, void* d_ws, size_t ws_size,
                              hipStream_t stream) {
  (void)in_sizes; (void)n_in; (void)out_size; (void)ws_size;

  const float* xA      = (const float*)d_in[0];   // [1,T,128]
  const float* xB      = (const float*)d_in[1];   // [1,T,128]
  const float* img     = (const float*)d_in[2];   // [1,T,65]
  const float* PA      = (const float*)d_in[3];   // [T,12,128]
  const float* PB      = (const float*)d_in[4];   // [T,12,128]
  const int*   lenA    = (const int*)d_in[5];     // [T]
  const int*   lenB    = (const int*)d_in[6];     // [T]
  const float* Wih_vad = (const float*)d_in[7];   // [256,128]
  const float* Whh_vad = (const float*)d_in[8];   // [256,64]
  const float* b_vad   = (const float*)d_in[9];   // [256]
  const float* Wih_img = (const float*)d_in[10];  // [256,65]
  const float* Whh_img = (const float*)d_in[11];  // [256,64]
  const float* b_img   = (const float*)d_in[12];  // [256]
  const float* Wih_lng = (const float*)d_in[13];  // [256,128]
  const float* Whh_lng = (const float*)d_in[14];  // [256,64]
  const float* b_lng   = (const float*)d_in[15];  // [256]
  const float* fc_w    = (const float*)d_in[16];  // [1,320]
  const float* fc_b    = (const float*)d_in[17];  // [1]
  float* out = (float*)d_out;                     // [1,T,1]

  // workspace carve-up (floats)
  float* ws = (float*)d_ws;
  size_t off = 0;
  float* GvA = ws + off; off += (size_t)T_STEPS * G4H;          //  8 MB
  float* GvB = ws + off; off += (size_t)T_STEPS * G4H;          //  8 MB
  float* Gim = ws + off; off += (size_t)T_STEPS * G4H;          //  8 MB
  float* GlA = ws + off; off += (size_t)T_STEPS * LSEQ * G4H;   // 96 MB
  float* GlB = ws + off; off += (size_t)T_STEPS * LSEQ * G4H;   // 96 MB
  float* hA  = ws + off; off += (size_t)T_STEPS * HID;
  float* hB  = ws + off; off += (size_t)T_STEPS * HID;
  float* hI  = ws + off; off += (size_t)T_STEPS * HID;
  float* hpa = ws + off; off += (size_t)T_STEPS * HID;
  float* hpb = ws + off; off += (size_t)T_STEPS * HID;

  // Stage 1: batched input projections (WMMA fp32).
  // Grid: (M/16 M-tiles, 4 macro N-tiles of 64 cols), one wave per block.
  {
    dim3 blk(32, 1, 1);
    dim3 grd_dense(T_STEPS / 16, 4, 1);
    dim3 grd_lng((T_STEPS * LSEQ) / 16, 4, 1);
    proj_wmma_f32<128><<<grd_dense, blk, 0, stream>>>(xA,  Wih_vad, b_vad, GvA, T_STEPS);
    proj_wmma_f32<128><<<grd_dense, blk, 0, stream>>>(xB,  Wih_vad, b_vad, GvB, T_STEPS);
    proj_wmma_f32<65> <<<grd_dense, blk, 0, stream>>>(img, Wih_img, b_img, Gim, T_STEPS);
    proj_wmma_f32<128><<<grd_lng,   blk, 0, stream>>>(PA,  Wih_lng, b_lng, GlA, T_STEPS * LSEQ);
    proj_wmma_f32<128><<<grd_lng,   blk, 0, stream>>>(PB,  Wih_lng, b_lng, GlB, T_STEPS * LSEQ);
  }

  // Stage 2: five serial recurrences, one block each
  lstm_recur<<<5, 256, 0, stream>>>(GvA, GvB, Gim, GlA, GlB,
                                    Whh_vad, Whh_img, Whh_lng,
                                    lenA, lenB, hA, hB, hI, hpa, hpb);

  // Stage 3: FC + sigmoid
  fc_sigmoid<<<T_STEPS / 256, 256, 0, stream>>>(hA, hB, hI, hpa, hpb, fc_w, fc_b, out);
}